// DWT3D_51238959841296
// MI455X (gfx1250) — compile-verified
//
#include <hip/hip_runtime.h>
#include <stdint.h>

// Problem dims (fixed by the reference):  video [B,C,F,H,W] f32
static constexpr int B  = 4, C = 3, F = 16, H = 256, W = 256;
static constexpr int BC = B * C;                 // 12
static constexpr int F2 = F / 2;                 // 8
static constexpr int H2 = H / 2;                 // 128
static constexpr int W2 = W / 2;                 // 128
static constexpr size_t SUB_ELEMS = (size_t)BC * F2 * H2 * W2;  // 1,572,864 per subband

#define INV_SQRT2 0.70710678118654752440f

// One block = one (bc, f_out, h_out pair) tile over the full W row.
// 256 threads stage 8 input rows (2 frames x 2 heights x 2 h_out x 256 f32 = 8KB)
// into LDS with gfx1250 async global->LDS b128 copies (2 in flight per lane),
// then each lane computes 2 subbands x float2 for each of the 2 output rows.
__global__ __launch_bounds__(256) void haar3d_l1_kernel(const float* __restrict__ in,
                                                        float* __restrict__ out) {
    __shared__ float tile[8 * W];   // 8 rows x 256 f32 = 8 KB

    const int tid = threadIdx.x;
    const int bid = blockIdx.x;             // [0, BC*F2*(H2/2))
    const int hp  = bid % (H2 / 2);         // h_out pair index: ho = 2*hp + j
    const int t1i = bid / (H2 / 2);
    const int fo  = t1i % F2;
    const int bc  = t1i / F2;

    // ---------------- async stage: 8 rows -> LDS ----------------
    // row r: (frame sub fr, height sub hr); lane covers 16B = 4 floats of W.
    const int r   = tid >> 6;               // 0..3
    const int fr  = r >> 1;
    const int hr  = r & 1;
    const int col = (tid & 63) * 4;

    const size_t gbase = ((size_t)bc * F + (2 * fo + fr)) * H;

#pragma unroll
    for (int j = 0; j < 2; ++j) {
        const size_t gofs = (gbase + (4 * hp + 2 * j + hr)) * (size_t)W + col;
        const float* gptr = in + gofs;
        // Low 32 bits of the flat (generic) LDS pointer == wave-relative LDS
        // byte address (flat aperture maps LDS via addr[31:0]).
        unsigned lds_addr = (unsigned)(uintptr_t)(&tile[(j * 4 + r) * W + col]);
        asm volatile("global_load_async_to_lds_b128 %0, %1, off th:TH_LOAD_NT"
                     :: "v"(lds_addr), "v"(gptr)
                     : "memory");
    }
    asm volatile("s_wait_asynccnt 0" ::: "memory");
    __syncthreads();

    // ---------------- butterfly compute ----------------
    // g selects which frame/height combination this lane produces:
    //   g=0 -> aa (LLL/LLH), g=1 -> ad (LHL/LHH), g=2 -> da (HLL/HLH), g=3 -> dd (HHL/HHH)
    const int g  = tid >> 6;
    const int q  = tid & 63;                // output columns wo = 2q, 2q+1
    const int c4 = 4 * q;

    const float k  = INV_SQRT2;
    const float sf = (g < 2) ? 1.0f : -1.0f;   // frame low (a) vs high (d)
    const float sh = (g & 1) ? -1.0f : 1.0f;   // height low vs high

#pragma unroll
    for (int j = 0; j < 2; ++j) {
        const int base = (j * 4) * W + c4;
        const float4 x00 = *(const float4*)&tile[base + 0 * W];   // (f0, h0)
        const float4 x01 = *(const float4*)&tile[base + 1 * W];   // (f0, h1)
        const float4 x10 = *(const float4*)&tile[base + 2 * W];   // (f1, h0)
        const float4 x11 = *(const float4*)&tile[base + 3 * W];   // (f1, h1)

        // frame stage (scaled by k), componentwise
        float t0x = (x00.x + sf * x10.x) * k;
        float t0y = (x00.y + sf * x10.y) * k;
        float t0z = (x00.z + sf * x10.z) * k;
        float t0w = (x00.w + sf * x10.w) * k;
        float t1x = (x01.x + sf * x11.x) * k;
        float t1y = (x01.y + sf * x11.y) * k;
        float t1z = (x01.z + sf * x11.z) * k;
        float t1w = (x01.w + sf * x11.w) * k;

        // height stage
        float yx = (t0x + sh * t1x) * k;
        float yy = (t0y + sh * t1y) * k;
        float yz = (t0z + sh * t1z) * k;
        float yw = (t0w + sh * t1w) * k;

        // width stage: two output columns
        float lo0 = (yx + yy) * k;
        float hi0 = (yx - yy) * k;
        float lo1 = (yz + yw) * k;
        float hi1 = (yz - yw) * k;

        // ---------------- non-temporal streamed stores ----------------
        const int ho = 2 * hp + j;
        const size_t oofs = (((size_t)bc * F2 + fo) * H2 + ho) * (size_t)W2 + 2 * q;
        float* outLo = out + (size_t)(2 * g)     * SUB_ELEMS + oofs;
        float* outHi = out + (size_t)(2 * g + 1) * SUB_ELEMS + oofs;

        union { float2 v; double d; } uLo, uHi;
        uLo.v = make_float2(lo0, lo1);
        uHi.v = make_float2(hi0, hi1);
        __builtin_nontemporal_store(uLo.d, (double*)outLo);
        __builtin_nontemporal_store(uHi.d, (double*)outHi);
    }
}

extern "C" void kernel_launch(void* const* d_in, const int* in_sizes, int n_in,
                              void* d_out, int out_size, void* d_ws, size_t ws_size,
                              hipStream_t stream) {
    const float* video = (const float*)d_in[0];
    float* out = (float*)d_out;

    dim3 grid(BC * F2 * (H2 / 2));   // 12 * 8 * 64 = 6144 workgroups
    dim3 block(256);                 // 8 wave32
    haar3d_l1_kernel<<<grid, block, 0, stream>>>(video, out);
}